// LIF_77859167142119
// MI455X (gfx1250) — compile-verified
//
#include <hip/hip_runtime.h>

// LIF neuron forward for MI455X (gfx1250, wave32).
// Phase 1: bandwidth-bound GEMV (100.7 MB reads @ 23.3 TB/s ~ 4.3 us) via
//          V_WMMA_F32_16X16X4_F32 — f32 kept since roofline is HBM, not MATH.
//          K loop fully unrolled (D=128 compile-time) -> 32 chained WMMAs.
// Phase 2: per-row sequential scan with rolling zero-window state.

typedef float v2f __attribute__((ext_vector_type(2)));
typedef float v8f __attribute__((ext_vector_type(8)));

#define T_STEPS 768
#define D_DIM   128
#define V_TH    1.27f
#define DECAY_F 0.02f   // 1/50

// ---------------------------------------------------------------------------
// GEMV: ode0[r] = dot(inputs[r, 0:D], W) + bias, r = b*T + t, stored
// transposed into odeT[t*B + b] so the scan reads coalesced.
// One wave handles 16 consecutive rows with 32 chained f32 WMMAs (K=4 each).
// A layout (16x4 f32): lanes 0-15 -> M=lane, {VGPR0,VGPR1} = {K0,K1};
//                      lanes 16-31 -> M=lane-16, {K2,K3}.
// B (4x16) = W broadcast along N; D column N=0: lane 0 holds rows M=0..7 in
// c[0..7], lane 16 holds rows M=8..15.
// ---------------------------------------------------------------------------
__global__ __launch_bounds__(256) void lif_gemv_wmma(
    const float* __restrict__ in, const float* __restrict__ W,
    const float* __restrict__ bias, float* __restrict__ odeT,
    long R, int Bn) {
  const int lane = threadIdx.x & 31;
  const int wave = threadIdx.x >> 5;
  const long gw = (long)blockIdx.x * 8 + wave;
  const long row_base = gw * 16;
  if (row_base >= R) return;                 // wave-uniform: EXEC stays all-1s

  const int half = lane >> 4;                // 0: lanes 0-15, 1: lanes 16-31
  const int m = lane & 15;
  const int aoff = half ? 2 : 0;             // A K-pair offset for this lane
  const int w0 = half ? 2 : 0;               // B K->lane-half mapping
  const int w1 = half ? 3 : 1;
  const float* __restrict__ arp = in + (row_base + m) * (long)D_DIM;

  v8f c = {};
#pragma unroll
  for (int k = 0; k < D_DIM; k += 4) {
    // A: this lane's 2 K-values for its row (8B aligned).
    v2f a = *(const v2f*)(arp + k + aoff);
    // B: W broadcast across all N columns for this K-chunk.
    v2f bv;
    bv.x = W[k + w0];
    bv.y = W[k + w1];
    c = __builtin_amdgcn_wmma_f32_16x16x4_f32(
        /*neg_a=*/false, a, /*neg_b=*/false, bv,
        /*c_mod=*/(short)0, c, /*reuse_a=*/false, /*reuse_b=*/false);
  }

  if (m == 0) {                              // lanes 0 and 16 own column N=0
    const float bs = bias[0];
#pragma unroll
    for (int j = 0; j < 8; ++j) {
      long r = row_base + half * 8 + j;
      int bidx = (int)(r / T_STEPS);
      int t = (int)(r % T_STEPS);
      odeT[(long)t * Bn + bidx] = c[j] + bs;
    }
  }
}

// ---------------------------------------------------------------------------
// Sequential LIF scan. One thread per batch row b; loads odeT[t*B+b] are
// coalesced (one 128B line per wave per step, buffer is L2-resident).
// Spike at step s>=1 zeroes ode[s-1 .. s+5]:
//   read at t uses 0 iff t <= zero_until(spikes < t)
//   final ode[t-1] is 0 iff (t-1) <= zero_until(spikes <= t)  (delayed write)
// Outputs written flat into d_out as [v | s | ode | ode], each [B,T].
// ---------------------------------------------------------------------------
__global__ void lif_scan(const float* __restrict__ odeT,
                         float* __restrict__ out, int Bn) {
  const int b = blockIdx.x * blockDim.x + threadIdx.x;
  if (b >= Bn) return;
  const long BT = (long)Bn * T_STEPS;
  float* __restrict__ out_v  = out;
  float* __restrict__ out_s  = out + BT;
  float* __restrict__ out_o1 = out + 2 * BT;
  float* __restrict__ out_o2 = out + 3 * BT;
  const long ob = (long)b * T_STEPS;

  float v = 0.0f;        // V_REST
  int counter = 0;
  int zero_until = -1;   // max(spike_t + 5) over past spikes with spike_t >= 1
  float prev_raw = 0.0f;

  for (int t = 0; t < T_STEPS; ++t) {
    const float raw = odeT[(long)t * Bn + b];
    v *= (1.0f - DECAY_F);                   // leak toward V_REST = 0
    if (counter == 1) v = 0.0f;              // V_RESET
    const float used = (t <= zero_until) ? 0.0f : raw;   // zeroing by past spikes
    v += used;
    const float s = (v - V_TH > 0.0f) ? 1.0f : 0.0f;
    counter += 2 * (int)s;                   // (SPIKE_DUR + 1) * s
    if (s > 0.0f && t >= 1) {                // t=0 spike has empty zero window
      const int zu = t + 5;
      if (zu > zero_until) zero_until = zu;
    }
    counter -= 1; if (counter < 0) counter = 0;

    out_v[ob + t] = v;
    out_s[ob + t] = s;
    if (t >= 1) {                            // delayed: now zero_until covers s<=t
      const float op = ((t - 1) <= zero_until) ? 0.0f : prev_raw;
      out_o1[ob + (t - 1)] = op;
      out_o2[ob + (t - 1)] = op;
    }
    prev_raw = raw;
  }
  const float ol = ((T_STEPS - 1) <= zero_until) ? 0.0f : prev_raw;
  out_o1[ob + (T_STEPS - 1)] = ol;
  out_o2[ob + (T_STEPS - 1)] = ol;
}

extern "C" void kernel_launch(void* const* d_in, const int* in_sizes, int n_in,
                              void* d_out, int out_size, void* d_ws, size_t ws_size,
                              hipStream_t stream) {
  const float* in   = (const float*)d_in[0];   // [B, T, D] f32
  const float* W    = (const float*)d_in[1];   // [D, 1]    f32
  const float* bias = (const float*)d_in[2];   // [1]       f32
  float* out = (float*)d_out;                  // [4 * B * T] f32

  const int  D  = in_sizes[1];                 // 128 (compile-time specialized)
  const long R  = (long)in_sizes[0] / D;       // B*T = 196608
  const int  Bn = (int)(R / T_STEPS);          // 256

  float* odeT = (float*)d_ws;                  // [T, B] f32 scratch (786 KB)

  const long waves  = (R + 15) / 16;           // 16 rows per wave
  const int  blocks = (int)((waves + 7) / 8);  // 8 waves per 256-thread block
  lif_gemv_wmma<<<blocks, 256, 0, stream>>>(in, W, bias, odeT, R, Bn);
  lif_scan<<<(Bn + 31) / 32, 32, 0, stream>>>(odeT, out, Bn);
}